// LightGCN_72962904424576
// MI455X (gfx1250) — compile-verified
//
#include <hip/hip_runtime.h>
#include <hip/hip_bf16.h>

#define NUM_USERS 100000
#define NUM_ITEMS 50000
#define N_NODES   150000
#define N_EDGES   2000000
#define D         128
#define NL        4
#define BATCH     8192

typedef __attribute__((ext_vector_type(2))) float v2f;
typedef __attribute__((ext_vector_type(8))) float v8f;

__device__ inline float wave_sum(float v) {
    #pragma unroll
    for (int off = 16; off >= 1; off >>= 1) v += __shfl_xor(v, off, 32);
    return v;
}

// ---------------- zero a float buffer (float4 granularity) ----------------
__global__ void k_zero(float* __restrict__ p, long long n4) {
    long long i = (long long)blockIdx.x * blockDim.x + threadIdx.x;
    long long st = (long long)gridDim.x * blockDim.x;
    float4 z; z.x = z.y = z.z = z.w = 0.f;
    for (; i < n4; i += st) ((float4*)p)[i] = z;
}

// ---------------- x = acc = concat(emb_user, emb_item) ----------------
__global__ void k_init(const float* __restrict__ eu, const float* __restrict__ ei,
                       float* __restrict__ x, float* __restrict__ acc) {
    const long long n4 = (long long)N_NODES * (D / 4);
    const long long uEnd = (long long)NUM_USERS * (D / 4);
    long long i = (long long)blockIdx.x * blockDim.x + threadIdx.x;
    long long st = (long long)gridDim.x * blockDim.x;
    for (; i < n4; i += st) {
        float4 v = (i < uEnd) ? ((const float4*)eu)[i] : ((const float4*)ei)[i - uEnd];
        ((float4*)x)[i] = v;
        ((float4*)acc)[i] = v;
    }
}

// ---------------- one SpMM layer: xn[row] += val * x[col] ----------------
// one wave32 per edge; lane handles 4 consecutive dims (float4 gather, f32 atomics)
__global__ void k_spmm(const int* __restrict__ erow, const int* __restrict__ ecol,
                       const float* __restrict__ eval, const float* __restrict__ x,
                       float* __restrict__ xn, int nEdges) {
    int lane = threadIdx.x & 31;
    long long edge = ((long long)blockIdx.x * blockDim.x + threadIdx.x) >> 5;
    if (edge >= nEdges) return;
    int row = erow[edge], col = ecol[edge];
    float v = eval[edge];
    const float4 xc = *(const float4*)(x + (size_t)col * D + lane * 4);
    float* dst = xn + (size_t)row * D + lane * 4;
    unsafeAtomicAdd(dst + 0, v * xc.x);
    unsafeAtomicAdd(dst + 1, v * xc.y);
    unsafeAtomicAdd(dst + 2, v * xc.z);
    unsafeAtomicAdd(dst + 3, v * xc.w);
}

// ---------------- acc += xn ----------------
__global__ void k_accum(float* __restrict__ acc, const float* __restrict__ xn, long long n4) {
    long long i = (long long)blockIdx.x * blockDim.x + threadIdx.x;
    long long st = (long long)gridDim.x * blockDim.x;
    for (; i < n4; i += st) {
        float4 a = ((float4*)acc)[i];
        float4 b = ((const float4*)xn)[i];
        a.x += b.x; a.y += b.y; a.z += b.z; a.w += b.w;
        ((float4*)acc)[i] = a;
    }
}

// ---------------- column sums of acc split by user/item rows ----------------
// 128 threads/block, thread d owns dim d; 128 rows per block, 2 atomics per thread
__global__ void k_nodemean(const float* __restrict__ acc, float* __restrict__ gsum) {
    int d = threadIdx.x;
    int r0 = blockIdx.x * 128;
    float su = 0.f, si = 0.f;
    for (int k = 0; k < 128; ++k) {
        int n = r0 + k;
        if (n >= N_NODES) break;
        float v = acc[(size_t)n * D + d];
        if (n < NUM_USERS) su += v; else si += v;
    }
    unsafeAtomicAdd(&gsum[d], su);
    unsafeAtomicAdd(&gsum[D + d], si);
}

// ---------------- fused gather + LayerNorm + WMMA-f32 MLP + residual dot ----
// block = 256 threads (8 wave32s) -> 16 batch rows; wave w owns output cols [16w,16w+16)
__launch_bounds__(256)
__global__ void k_batch(const int* __restrict__ users, const int* __restrict__ items,
                        const float* __restrict__ acc,
                        const float* __restrict__ ubias, const float* __restrict__ ibias,
                        const float* __restrict__ gsum,
                        const float* __restrict__ gam, const float* __restrict__ bet,
                        const float* __restrict__ W1, const float* __restrict__ b1,
                        const float* __restrict__ W2, const float* __restrict__ b2,
                        float* __restrict__ out) {
    __shared__ float X[16][644];   // 640 cols, padded stride: conflict-free column reads
    __shared__ float H[16][132];
    __shared__ float dotS[16];

    const int tid = threadIdx.x;
    const int lane = tid & 31;
    const int wave = tid >> 5;
    const float inv5 = 0.2f;
    const float invNU = 1.0f / (float)NUM_USERS;
    const float invNI = 1.0f / (float)NUM_ITEMS;

    // ---- phase 1: build 5D-concat row, LayerNorm it, compute dot(u,i) ----
    #pragma unroll
    for (int rr = 0; rr < 2; ++rr) {
        int r = wave * 2 + rr;
        int gr = blockIdx.x * 16 + r;
        int u = users[gr], it = items[gr];
        int db = lane * 4;

        float4 u4 = *(const float4*)(acc + (size_t)u * D + db);
        float4 i4 = *(const float4*)(acc + ((size_t)NUM_USERS + it) * D + db);
        u4.x *= inv5; u4.y *= inv5; u4.z *= inv5; u4.w *= inv5;
        i4.x *= inv5; i4.y *= inv5; i4.z *= inv5; i4.w *= inv5;
        float4 ub = *(const float4*)(ubias + (size_t)u * D + db);
        float4 ib = *(const float4*)(ibias + (size_t)it * D + db);
        ub.x *= 0.5f; ub.y *= 0.5f; ub.z *= 0.5f; ub.w *= 0.5f;
        ib.x *= 0.5f; ib.y *= 0.5f; ib.z *= 0.5f; ib.w *= 0.5f;
        float4 g;
        g.x = (gsum[db + 0] * invNU + gsum[D + db + 0] * invNI) * inv5;
        g.y = (gsum[db + 1] * invNU + gsum[D + db + 1] * invNI) * inv5;
        g.z = (gsum[db + 2] * invNU + gsum[D + db + 2] * invNI) * inv5;
        g.w = (gsum[db + 3] * invNU + gsum[D + db + 3] * invNI) * inv5;

        X[r][      db + 0] = u4.x; X[r][      db + 1] = u4.y; X[r][      db + 2] = u4.z; X[r][      db + 3] = u4.w;
        X[r][128 + db + 0] = i4.x; X[r][128 + db + 1] = i4.y; X[r][128 + db + 2] = i4.z; X[r][128 + db + 3] = i4.w;
        X[r][256 + db + 0] = ub.x; X[r][256 + db + 1] = ub.y; X[r][256 + db + 2] = ub.z; X[r][256 + db + 3] = ub.w;
        X[r][384 + db + 0] = ib.x; X[r][384 + db + 1] = ib.y; X[r][384 + db + 2] = ib.z; X[r][384 + db + 3] = ib.w;
        X[r][512 + db + 0] = g.x;  X[r][512 + db + 1] = g.y;  X[r][512 + db + 2] = g.z;  X[r][512 + db + 3] = g.w;

        float s  = u4.x + u4.y + u4.z + u4.w + i4.x + i4.y + i4.z + i4.w
                 + ub.x + ub.y + ub.z + ub.w + ib.x + ib.y + ib.z + ib.w
                 + g.x + g.y + g.z + g.w;
        float sq = u4.x*u4.x + u4.y*u4.y + u4.z*u4.z + u4.w*u4.w
                 + i4.x*i4.x + i4.y*i4.y + i4.z*i4.z + i4.w*i4.w
                 + ub.x*ub.x + ub.y*ub.y + ub.z*ub.z + ub.w*ub.w
                 + ib.x*ib.x + ib.y*ib.y + ib.z*ib.z + ib.w*ib.w
                 + g.x*g.x + g.y*g.y + g.z*g.z + g.w*g.w;
        float dp = u4.x*i4.x + u4.y*i4.y + u4.z*i4.z + u4.w*i4.w;

        s  = wave_sum(s);
        sq = wave_sum(sq);
        dp = wave_sum(dp);
        if (lane == 0) dotS[r] = dp;

        float mu   = s * (1.0f / 640.0f);
        float var  = sq * (1.0f / 640.0f) - mu * mu;
        float rstd = rsqrtf(var + 1e-5f);
        #pragma unroll
        for (int sg = 0; sg < 5; ++sg) {
            #pragma unroll
            for (int k = 0; k < 4; ++k) {
                int col = sg * 128 + db + k;
                float v = X[r][col];
                X[r][col] = (v - mu) * rstd * gam[col] + bet[col];
            }
        }
    }
    __syncthreads();

    // ---- phase 2: [16x640]@[640x16] per wave via v_wmma_f32_16x16x4_f32 ----
    const int cb = wave * 16;
    const int m  = lane & 15;          // A row / B col within tile
    const int kh = (lane >> 4) * 2;    // K sub-slot: lanes 0-15 -> K0/K1, 16-31 -> K2/K3
    v8f c = {0.f, 0.f, 0.f, 0.f, 0.f, 0.f, 0.f, 0.f};
    for (int k0 = 0; k0 < 640; k0 += 4) {
        v2f a, b;
        a.x = X[m][k0 + kh];
        a.y = X[m][k0 + kh + 1];
        b.x = W1[(size_t)(k0 + kh)     * 128 + cb + m];
        b.y = W1[(size_t)(k0 + kh + 1) * 128 + cb + m];
        c = __builtin_amdgcn_wmma_f32_16x16x4_f32(false, a, false, b, (short)0, c,
                                                  false, false);
    }
    #pragma unroll
    for (int r8 = 0; r8 < 8; ++r8) {
        int M   = r8 + ((lane >> 4) << 3);   // C/D layout: vgpr r -> M=r (lo half), M=8+r (hi half)
        int col = cb + m;
        float h = c[r8] + b1[col];
        H[M][col] = h > 0.f ? h : 0.f;
    }
    __syncthreads();

    // ---- phase 3: h @ W2 + b2, residual blend ----
    if (tid < 16) {
        int gr = blockIdx.x * 16 + tid;
        float sc = b2[0];
        for (int d2 = 0; d2 < 128; ++d2) sc += H[tid][d2] * W2[d2];
        out[gr] = 0.3f * dotS[tid] + 0.7f * sc;
    }
}

extern "C" void kernel_launch(void* const* d_in, const int* in_sizes, int n_in,
                              void* d_out, int out_size, void* d_ws, size_t ws_size,
                              hipStream_t stream) {
    const int*   users = (const int*)d_in[0];
    const int*   items = (const int*)d_in[1];
    const int*   erow  = (const int*)d_in[2];
    const int*   ecol  = (const int*)d_in[3];
    const float* eval  = (const float*)d_in[4];
    const float* emb_u = (const float*)d_in[5];
    const float* emb_i = (const float*)d_in[6];
    const float* ubias = (const float*)d_in[7];
    const float* ibias = (const float*)d_in[8];
    const float* gam   = (const float*)d_in[9];
    const float* bet   = (const float*)d_in[10];
    const float* W1    = (const float*)d_in[11];
    const float* b1    = (const float*)d_in[12];
    const float* W2    = (const float*)d_in[13];
    const float* b2    = (const float*)d_in[14];
    float* out = (float*)d_out;

    const size_t ND = (size_t)N_NODES * D;   // 19.2M floats (76.8 MB)
    float* bufA = (float*)d_ws;
    float* bufB = bufA + ND;
    float* acc  = bufB + ND;
    float* gsum = acc + ND;                  // 256 floats: [0..127]=user colsum, [128..255]=item

    const long long n4 = (long long)ND / 4;
    k_zero<<<1, 64, 0, stream>>>(gsum, 64);
    k_init<<<18750, 256, 0, stream>>>(emb_u, emb_i, bufA, acc);

    float* cur = bufA;
    float* nxt = bufB;
    for (int l = 0; l < NL; ++l) {
        k_zero<<<18750, 256, 0, stream>>>(nxt, n4);
        k_spmm<<<(N_EDGES * 32) / 256, 256, 0, stream>>>(erow, ecol, eval, cur, nxt, N_EDGES);
        k_accum<<<18750, 256, 0, stream>>>(acc, nxt, n4);
        float* t = cur; cur = nxt; nxt = t;
    }

    k_nodemean<<<(N_NODES + 127) / 128, 128, 0, stream>>>(acc, gsum);
    k_batch<<<BATCH / 16, 256, 0, stream>>>(users, items, acc, ubias, ibias, gsum,
                                            gam, bet, W1, b1, W2, b2, out);
}